// PhasedTorchRNN_52682068853083
// MI455X (gfx1250) — compile-verified
//
#include <hip/hip_runtime.h>
#include <cmath>

typedef __attribute__((ext_vector_type(2))) float v2f;
typedef __attribute__((ext_vector_type(8))) float v8f;

__device__ __forceinline__ float tanh_fast(float x) {
#if __has_builtin(__builtin_amdgcn_tanhf)
  return __builtin_amdgcn_tanhf(x);
#elif __has_builtin(__builtin_amdgcn_tanh_f32)
  return __builtin_amdgcn_tanh_f32(x);
#else
  return tanhf(x);
#endif
}

// One wave (32 lanes) advances 16 independent sequences with f32 WMMA.
// A-layout (f32 16x4): lane = m + 16*(k>=2), VGPR v holds k = 2*(lane>=16)+v.
// B-layout (mirror):   VGPR v, lanes0-15 -> K=v, lanes16-31 -> K=v+2, N=lane%16.
// C/D layout: VGPR r holds row M=r (lanes 0-15, N=lane) and M=r+8 (lanes 16-31).
__global__ __launch_bounds__(32)
void rnn2_wmma_kernel(const int* __restrict__ tokens,   // [B,T]
                      const int* __restrict__ lengths,  // [B]
                      const float* __restrict__ Wih0,   // [4,2]
                      const float* __restrict__ Whh0,   // [4,4]
                      const float* __restrict__ bih0,   // [4]
                      const float* __restrict__ bhh0,   // [4]
                      const float* __restrict__ Wih1,   // [4,4]
                      const float* __restrict__ Whh1,   // [4,4]
                      const float* __restrict__ bih1,   // [4]
                      const float* __restrict__ bhh1,   // [4]
                      const float* __restrict__ h0,     // [2,4]
                      float* __restrict__ out,          // [B,4]
                      int Bn, int T)
{
  // [0,64):   scr1 (layer-1 transpose, scr[j*16+m])
  // [64,128): scr2 (layer-2 transpose)
  // [128,384): per-lane dump slots for lanes with m>=4 (branchless stores)
  __shared__ float scr[384];

  const int lane = threadIdx.x;       // 0..31
  const int m    = lane & 15;         // tile row (batch within tile) / B col
  const int hi   = lane >> 4;         // lane half
  const int k0   = hi * 2;            // K pair base
  const int base = blockIdx.x * 16;
  const int b    = min(base + m, Bn - 1);

  // ---- per-lane fused constants ----
  // layer1 addend for unit j: Wih0[j][tok] + bih0[j] + bhh0[j], j in {k0, k0+1}
  const float bs0a = bih0[k0]     + bhh0[k0];
  const float bs0b = bih0[k0 + 1] + bhh0[k0 + 1];
  const float c0a = Wih0[(k0)     * 2 + 0] + bs0a;
  const float c1a = Wih0[(k0)     * 2 + 1] + bs0a;
  const float c0b = Wih0[(k0 + 1) * 2 + 0] + bs0b;
  const float c1b = Wih0[(k0 + 1) * 2 + 1] + bs0b;

  // ---- B operands: W^T zero-padded to 4x16; layer-2 bias folded into C ----
  v2f w0b = {0.0f, 0.0f}, wib = {0.0f, 0.0f}, whb = {0.0f, 0.0f};
  v8f cbias = {0,0,0,0,0,0,0,0};
  const bool n4 = (m < 4);
  if (n4) {
    w0b[0] = Whh0[m * 4 + k0];  w0b[1] = Whh0[m * 4 + k0 + 1];
    wib[0] = Wih1[m * 4 + k0];  wib[1] = Wih1[m * 4 + k0 + 1];
    whb[0] = Whh1[m * 4 + k0];  whb[1] = Whh1[m * 4 + k0 + 1];
    const float bb = bih1[m] + bhh1[m];
#pragma unroll
    for (int r = 0; r < 8; ++r) cbias[r] = bb;
  }
  const v8f cz = {0,0,0,0,0,0,0,0};

  // ---- recurrent state in A layout ----
  v2f h1 = { h0[k0],     h0[k0 + 1]     };
  v2f h2 = { h0[4 + k0], h0[4 + k0 + 1] };

  const int len = lengths[b];
  int lmax = len;
#pragma unroll
  for (int off = 16; off > 0; off >>= 1)
    lmax = max(lmax, __shfl_xor(lmax, off, 32));
  lmax = __builtin_amdgcn_readfirstlane(lmax);   // force SGPR -> scalar loop
  const int tmax = (lmax + 3) & ~3;              // uniform, <= T

  const int* rowp = tokens + (size_t)b * (size_t)T;

  // Branchless transpose-store pointers: real slot for m<4, dump otherwise.
  float* st1 = n4 ? &scr[m * 16 + hi * 8]      : &scr[128 + lane * 8];
  float* st2 = n4 ? &scr[64 + m * 16 + hi * 8] : &scr[128 + lane * 8];
  const float* rd1 = &scr[k0 * 16 + m];
  const float* rd2 = &scr[64 + k0 * 16 + m];

  for (int t0 = 0; t0 < tmax; t0 += 4) {
    const int4 tk = *(const int4*)(rowp + t0);
    __builtin_prefetch(rowp + t0 + 64, 0, 1);   // speculative; OOB is dropped
    const int tks[4] = { tk.x, tk.y, tk.z, tk.w };
#pragma unroll
    for (int u = 0; u < 4; ++u) {
      const int t   = t0 + u;
      const int tok = tks[u];

      // ===== layer 1: D = h1(16x4) @ Whh0^T(4x16) =====
      v8f d1 = __builtin_amdgcn_wmma_f32_16x16x4_f32(
          false, h1, false, w0b, (short)0, cz, false, false);
      ((float4*)st1)[0] = make_float4(d1[0], d1[1], d1[2], d1[3]);
      ((float4*)st1)[1] = make_float4(d1[4], d1[5], d1[6], d1[7]);
      const float p0 = rd1[0];
      const float p1 = rd1[16];
      const float n0 = tanh_fast(p0 + (tok ? c1a : c0a));
      const float n1 = tanh_fast(p1 + (tok ? c1b : c0b));
      const bool valid = (t < len);
      h1[0] = valid ? n0 : h1[0];
      h1[1] = valid ? n1 : h1[1];

      // ===== layer 2: D = a1 @ Wih1^T + h2 @ Whh1^T + bias2 =====
      v8f dA = __builtin_amdgcn_wmma_f32_16x16x4_f32(
          false, h1, false, wib, (short)0, cbias, false, false);
      v8f d2 = __builtin_amdgcn_wmma_f32_16x16x4_f32(
          false, h2, false, whb, (short)0, dA, false, false);
      ((float4*)st2)[0] = make_float4(d2[0], d2[1], d2[2], d2[3]);
      ((float4*)st2)[1] = make_float4(d2[4], d2[5], d2[6], d2[7]);
      const float q0 = rd2[0];
      const float q1 = rd2[16];
      const float m0 = tanh_fast(q0);
      const float m1 = tanh_fast(q1);
      h2[0] = valid ? m0 : h2[0];
      h2[1] = valid ? m1 : h2[1];
    }
  }

  // final layer-2 hidden state; lane halves cover disjoint j = {k0, k0+1}
  out[(size_t)b * 4 + k0 + 0] = h2[0];
  out[(size_t)b * 4 + k0 + 1] = h2[1];
}

extern "C" void kernel_launch(void* const* d_in, const int* in_sizes, int n_in,
                              void* d_out, int out_size, void* d_ws, size_t ws_size,
                              hipStream_t stream) {
  const int*   tokens  = (const int*)  d_in[0];
  const int*   lengths = (const int*)  d_in[1];
  const float* Wih0    = (const float*)d_in[2];
  const float* Whh0    = (const float*)d_in[3];
  const float* bih0    = (const float*)d_in[4];
  const float* bhh0    = (const float*)d_in[5];
  const float* Wih1    = (const float*)d_in[6];
  const float* Whh1    = (const float*)d_in[7];
  const float* bih1    = (const float*)d_in[8];
  const float* bhh1    = (const float*)d_in[9];
  const float* h0      = (const float*)d_in[10];

  const int Bn = in_sizes[1];             // lengths: [B]
  const int T  = in_sizes[0] / Bn;        // tokens:  [B,T]
  const int nblocks = (Bn + 15) / 16;     // 16 sequences per wave

  hipLaunchKernelGGL(rnn2_wmma_kernel, dim3(nblocks), dim3(32), 0, stream,
                     tokens, lengths, Wih0, Whh0, bih0, bhh0,
                     Wih1, Whh1, bih1, bhh1, h0,
                     (float*)d_out, Bn, T);
}